// ContrastiveLoss_12532714569938
// MI455X (gfx1250) — compile-verified
//
#include <hip/hip_runtime.h>

// InfoNCE contrastive loss, fused single-pass, MI455X (gfx1250, wave32).
// Memory-bound: 1.07 GB of negatives streamed once -> ~46us floor @ 23.3 TB/s.
// Dot products and squared norms computed on the matrix pipe via
// V_WMMA_F32_16X16X4_F32:  C1 += A x broadcast(a_hat)  (dots in every column)
//                          C2 += A x A                 (norms^2 on the diagonal,
//                              since the A-fragment layout == B-fragment of A^T).
// Staging uses GLOBAL_LOAD_ASYNC_TO_LDS_B128 (ASYNCcnt) — confirmed present on
// this toolchain; typed as (v4i AS1*, v4i AS3*, imm, imm).

typedef float v2f __attribute__((ext_vector_type(2)));
typedef float v4f __attribute__((ext_vector_type(4)));
typedef float v8f __attribute__((ext_vector_type(8)));
typedef int   v4i __attribute__((ext_vector_type(4)));

#define DDIM        128
#define NNEG        256
#define TILE_ROWS   64
#define ROW_STRIDE  132      // 128 + 4 pad -> conflict-free A-fragment ds reads
#define TEMP        0.5f
#define NEPS        1e-12f

#if defined(__has_builtin)
#if __has_builtin(__builtin_amdgcn_global_load_async_to_lds_b128)
#define HAVE_ASYNC_LDS 1
#endif
#if __has_builtin(__builtin_amdgcn_s_wait_asynccnt)
#define HAVE_WAIT_ASYNC_BUILTIN 1
#endif
#endif

typedef __attribute__((address_space(1))) v4i* gptr_v4i;
typedef __attribute__((address_space(3))) v4i* lptr_v4i;

__device__ __forceinline__ void wait_asynccnt0() {
#if defined(HAVE_WAIT_ASYNC_BUILTIN)
  __builtin_amdgcn_s_wait_asynccnt(0);
#else
  asm volatile("s_wait_asynccnt 0x0" ::: "memory");
#endif
}

// Branch-free extraction of c[ci], ci in [0,8) (lane-divergent) -> v_cndmask tree.
__device__ __forceinline__ float pick8(const v8f& c, int ci) {
  float s01 = (ci & 1) ? c[1] : c[0];
  float s23 = (ci & 1) ? c[3] : c[2];
  float s45 = (ci & 1) ? c[5] : c[4];
  float s67 = (ci & 1) ? c[7] : c[6];
  float a   = (ci & 2) ? s23 : s01;
  float b   = (ci & 2) ? s67 : s45;
  return (ci & 4) ? b : a;
}

__launch_bounds__(256)
__global__ void infonce_rows_kernel(const float* __restrict__ anchors,
                                    const float* __restrict__ positives,
                                    const float* __restrict__ negatives,
                                    const float* __restrict__ weights,
                                    float* __restrict__ row_loss) {
  __shared__ float tile[TILE_ROWS * ROW_STRIDE];   // 33,792 B
  __shared__ float a_hat[DDIM];
  __shared__ float dots[NNEG];
  __shared__ float nrm2[NNEG];
  __shared__ float red[256];

  const int b    = blockIdx.x;
  const int t    = threadIdx.x;
  const int lane = t & 31;
  const int wave = t >> 5;

  // ---------------- Phase 1: anchor/positive norms + pos similarity --------
  float av = 0.f, pv = 0.f;
  if (t < DDIM) {
    av = anchors  [(size_t)b * DDIM + t];
    pv = positives[(size_t)b * DDIM + t];
  }

  auto blockReduceSum = [&](float v) -> float {
    red[t] = v; __syncthreads();
#pragma unroll
    for (int s = 128; s > 0; s >>= 1) {
      if (t < s) red[t] += red[t + s];
      __syncthreads();
    }
    float r = red[0];
    __syncthreads();
    return r;
  };

  const float saa = blockReduceSum(av * av);
  const float spp = blockReduceSum(pv * pv);
  const float sap = blockReduceSum(av * pv);

  const float w         = weights[b];
  const float inv_na    = 1.f / fmaxf(sqrtf(saa), NEPS);
  const float inv_np    = 1.f / fmaxf(sqrtf(spp), NEPS);
  const float scale     = w / TEMP;
  const float pos_logit = sap * inv_na * inv_np * scale;

  if (t < DDIM) a_hat[t] = av * inv_na;
  __syncthreads();

  // ---------------- Phase 2: stream negatives, WMMA dots + norms -----------
  const float* base = negatives + (size_t)b * NNEG * DDIM;

  for (int tl = 0; tl < NNEG / TILE_ROWS; ++tl) {
    const float* src = base + (size_t)tl * TILE_ROWS * DDIM;

    // Coalesced staging: 64 rows x 128 fp32 (2048 x b128), direct-to-LDS async.
#pragma unroll
    for (int i = 0; i < (TILE_ROWS * DDIM / 4) / 256; ++i) {
      const int f   = t + 256 * i;   // float4 index
      const int row = f >> 5;        // 32 float4 per row
      const int c4  = f & 31;
#if defined(HAVE_ASYNC_LDS)
      __builtin_amdgcn_global_load_async_to_lds_b128(
          (gptr_v4i)(src + row * DDIM + c4 * 4),
          (lptr_v4i)&tile[row * ROW_STRIDE + c4 * 4],
          0, 0);
#else
      v4f val = *(const v4f*)(src + row * DDIM + c4 * 4);
      *(v4f*)&tile[row * ROW_STRIDE + c4 * 4] = val;
#endif
    }

    // Hint-prefetch next tile into cache (global_prefetch_b8).
    if (tl + 1 < NNEG / TILE_ROWS) {
      const float* nxt = src + TILE_ROWS * DDIM;
#pragma unroll
      for (int j = 0; j < 4; ++j)
        __builtin_prefetch(nxt + ((size_t)t * 4 + j) * 32, 0, 0);
    }
#if defined(HAVE_ASYNC_LDS)
    wait_asynccnt0();   // this wave's async copies are in LDS
#endif
    __syncthreads();

    // Waves 0..3 each own a 16-row group (compute is free; BW-bound kernel).
    if (wave < 4) {
      const int m    = lane & 15;
      const int koff = (lane < 16) ? 0 : 2;  // A layout: lanes 0-15 hold K=0,1; 16-31 hold K=2,3
      const float* arow = &tile[(wave * 16 + m) * ROW_STRIDE];

      v8f c1 = {};   // dots accumulator
      v8f c2 = {};   // Gram accumulator (diag = |n|^2)
#pragma unroll 4
      for (int k0 = 0; k0 < DDIM; k0 += 4) {
        v2f afrag = *(const v2f*)(arow  + k0 + koff);
        v2f bfrag = *(const v2f*)(a_hat + k0 + koff);  // B[k][j] = a_hat[k] for all j
        c1 = __builtin_amdgcn_wmma_f32_16x16x4_f32(
                 false, afrag, false, bfrag, (short)0, c1, false, false);
        c2 = __builtin_amdgcn_wmma_f32_16x16x4_f32(
                 false, afrag, false, afrag, (short)0, c2, false, false);
      }

      // Diagonal lives at lane==M==N: rows 0-7 on lanes 0-7 (c[lane]),
      // rows 8-15 on lanes 24-31 (c[lane-24]).  Branch-free select + masked store.
      const bool active = (lane < 8) || (lane >= 24);
      const int  ci     = (lane < 8) ? lane : (lane - 24);
      const int  g      = tl * TILE_ROWS + wave * 16 + (lane & 15);
      const float dval  = pick8(c1, ci);
      const float qval  = pick8(c2, ci);
      if (active) {
        dots[g] = dval;
        nrm2[g] = qval;
      }
    }
    __syncthreads();
  }

  // ---------------- Phase 3: logits -> logsumexp -> per-row loss -----------
  const float inv_nn = 1.f / fmaxf(sqrtf(nrm2[t]), NEPS);
  const float lneg   = dots[t] * inv_nn * scale;   // a_hat already normalized

  red[t] = lneg; __syncthreads();
#pragma unroll
  for (int s = 128; s > 0; s >>= 1) {
    if (t < s) red[t] = fmaxf(red[t], red[t + s]);
    __syncthreads();
  }
  const float mx = fmaxf(red[0], pos_logit);
  __syncthreads();

  float se = blockReduceSum(expf(lneg - mx));
  se += expf(pos_logit - mx);

  if (t == 0) row_loss[b] = (mx + logf(se)) - pos_logit;   // = -log_softmax[0]
}

// Deterministic final mean (fixed-order serial + tree; no float atomics).
__launch_bounds__(256)
__global__ void reduce_mean_kernel(const float* __restrict__ row_loss,
                                   float* __restrict__ out, int B) {
  __shared__ float red[256];
  const int t = threadIdx.x;
  float s = 0.f;
  for (int i = t; i < B; i += 256) s += row_loss[i];
  red[t] = s; __syncthreads();
#pragma unroll
  for (int k = 128; k > 0; k >>= 1) {
    if (t < k) red[t] += red[t + k];
    __syncthreads();
  }
  if (t == 0) out[0] = red[0] / (float)B;
}

extern "C" void kernel_launch(void* const* d_in, const int* in_sizes, int n_in,
                              void* d_out, int out_size, void* d_ws, size_t ws_size,
                              hipStream_t stream) {
  const float* anchors   = (const float*)d_in[0];
  const float* positives = (const float*)d_in[1];
  const float* negatives = (const float*)d_in[2];
  const float* weights   = (const float*)d_in[3];
  float* out = (float*)d_out;

  const int B = in_sizes[3];            // weights is [B]
  float* row_loss = (float*)d_ws;       // B floats of scratch

  infonce_rows_kernel<<<B, 256, 0, stream>>>(anchors, positives, negatives,
                                             weights, row_loss);
  reduce_mean_kernel<<<1, 256, 0, stream>>>(row_loss, out, B);
}